// GaussianTensor_69681549410424
// MI455X (gfx1250) — compile-verified
//
#include <hip/hip_runtime.h>

typedef __attribute__((ext_vector_type(2))) float v2f;
typedef __attribute__((ext_vector_type(8))) float v8f;

#define GSIZE 64
#define NTOT 65536
#define N_PER_BLOCK 64
#define LOG2PI_F 1.8378770664093453f

// ---------------------------------------------------------------------------
// Main kernel: accumulates  P[j] = Sum_n [ Sum_k m^2  - 2 * Sum_k m*x ]
// into out[j] via atomics.  Cross term via V_WMMA_F32_16X16X4_F32 (B columns
// replicated with x so every column of D accumulates the full dot product),
// square term via v2f packed-FMA accumulators fed straight from the B64 load
// register pairs (v_pk_fma_f32, no register shuffling).
// Block = 128 threads = 4 waves; wave w owns j-tile [16w, 16w+16).
// Block b owns n in [64b, 64b+64) -> streams a contiguous 1MB means slab.
// ---------------------------------------------------------------------------
__global__ __launch_bounds__(128) void gauss_wmma_main(
    const float* __restrict__ x, const float* __restrict__ m,
    float* __restrict__ out) {
  __shared__ float xs[N_PER_BLOCK * GSIZE];   // 16 KB x slab
  __shared__ float red_sq[4][32];
  __shared__ float red_cr[4][16];

  const int t    = threadIdx.x;
  const int wave = t >> 5;
  const int lane = t & 31;
  const int n0   = blockIdx.x * N_PER_BLOCK;

  // Stage x slab (coalesced float4 stream).
  {
    const float4* xsrc = (const float4*)(x + (size_t)n0 * GSIZE);
    float4*       xdst = (float4*)xs;
#pragma unroll
    for (int i = 0; i < (N_PER_BLOCK * GSIZE / 4) / 128; ++i)
      xdst[t + i * 128] = xsrc[t + i * 128];
  }
  __syncthreads();

  const int row = lane & 15;  // j within the 16-row tile (A-matrix M index)
  const int kh  = lane >> 4;  // lane-half selects k pair within chunk
  const int j0  = wave * 16;

  // A-operand base: lane holds m[n, j0+row, 4c + 2kh .. +1] per chunk c.
  const float* abase =
      m + (size_t)n0 * (GSIZE * GSIZE) + (size_t)(j0 + row) * GSIZE + 2 * kh;

  v8f c0 = {}, c1 = {}, c2 = {}, c3 = {};
  v2f sv0 = {}, sv1 = {}, sv2 = {}, sv3 = {};

  for (int nn = 0; nn < N_PER_BLOCK; ++nn) {
    const float* ap = abase + (size_t)nn * (GSIZE * GSIZE);
    const float* bp = xs + nn * GSIZE + 2 * kh;
    if (nn + 2 < N_PER_BLOCK)
      __builtin_prefetch(ap + 2 * (GSIZE * GSIZE), 0, 1);
#pragma unroll
    for (int cc = 0; cc < 16; ++cc) {
      v2f a = *(const v2f*)(ap + 4 * cc);  // A: k = 4cc+2kh, 4cc+2kh+1
      v2f b = *(const v2f*)(bp + 4 * cc);  // B: same k mapping, cols replicated
      // D = A x B + C ; every column of D accumulates Sum_k m[j,k]*x[k].
      // 4 rotating C accumulators + 4 rotating v2f sq accumulators for ILP;
      // acc += a*a contracts to a single v_pk_fma_f32 on the load pair.
      switch (cc & 3) {
        case 0:
          sv0 += a * a;
          c0 = __builtin_amdgcn_wmma_f32_16x16x4_f32(
              false, a, false, b, (short)0, c0, false, false);
          break;
        case 1:
          sv1 += a * a;
          c1 = __builtin_amdgcn_wmma_f32_16x16x4_f32(
              false, a, false, b, (short)0, c1, false, false);
          break;
        case 2:
          sv2 += a * a;
          c2 = __builtin_amdgcn_wmma_f32_16x16x4_f32(
              false, a, false, b, (short)0, c2, false, false);
          break;
        default:
          sv3 += a * a;
          c3 = __builtin_amdgcn_wmma_f32_16x16x4_f32(
              false, a, false, b, (short)0, c3, false, false);
          break;
      }
    }
  }
  v8f c = (c0 + c1) + (c2 + c3);
  v2f sv = (sv0 + sv1) + (sv2 + sv3);
  float sq = sv.x + sv.y;

  // Intra-wave recombine via LDS.
  red_sq[wave][lane] = sq;
  if (row == 0) {
    // C/D layout: vgpr r, lane-half h -> D[M = r + 8h] (all columns equal).
#pragma unroll
    for (int r = 0; r < 8; ++r) red_cr[wave][kh * 8 + r] = c[r];
  }
  __syncthreads();

  if (lane < 16) {
    // Sq_row[j]: lane j covers k%4 in {0,1}, lane j+16 covers {2,3}.
    float p = red_sq[wave][lane] + red_sq[wave][lane + 16] -
              2.0f * red_cr[wave][lane];
    atomicAdd(&out[j0 + lane], p);
  }
}

// ---------------------------------------------------------------------------
// S = Sum_{n,k} x^2  (16 MB stream, negligible)
// ---------------------------------------------------------------------------
__global__ __launch_bounds__(256) void gauss_sumsq(
    const float* __restrict__ x, float* __restrict__ s, int n) {
  __shared__ float red[256];
  v2f a01 = {}, a23 = {};
  const v2f* x2 = (const v2f*)x;
  for (int i = blockIdx.x * blockDim.x + threadIdx.x; i < n / 4;
       i += gridDim.x * blockDim.x) {
    v2f v0 = x2[2 * i];
    v2f v1 = x2[2 * i + 1];
    a01 += v0 * v0;
    a23 += v1 * v1;
  }
  v2f av = a01 + a23;
  red[threadIdx.x] = av.x + av.y;
  __syncthreads();
  for (int off = 128; off > 0; off >>= 1) {
    if (threadIdx.x < off) red[threadIdx.x] += red[threadIdx.x + off];
    __syncthreads();
  }
  if (threadIdx.x == 0) atomicAdd(s, red[0]);
}

// ---------------------------------------------------------------------------
// out[j] = -0.5 * (P[j] + S) - N * 32 * log(2*pi)
// ---------------------------------------------------------------------------
__global__ void gauss_finalize(float* __restrict__ out,
                               const float* __restrict__ s) {
  int j = threadIdx.x;
  if (j < GSIZE) {
    float v = out[j] + s[0];
    out[j] = -0.5f * v - (float)NTOT * 32.0f * LOG2PI_F;
  }
}

extern "C" void kernel_launch(void* const* d_in, const int* in_sizes, int n_in,
                              void* d_out, int out_size, void* d_ws,
                              size_t ws_size, hipStream_t stream) {
  const float* x = (const float*)d_in[0];   // (N, 64)
  const float* m = (const float*)d_in[1];   // (N, 64, 64)
  float* out = (float*)d_out;               // 64 floats
  float* ws  = (float*)d_ws;

  hipMemsetAsync(d_out, 0, (size_t)out_size * sizeof(float), stream);
  hipMemsetAsync(d_ws, 0, sizeof(float), stream);

  gauss_wmma_main<<<NTOT / N_PER_BLOCK, 128, 0, stream>>>(x, m, out);
  gauss_sumsq<<<256, 256, 0, stream>>>(x, ws, NTOT * GSIZE);
  gauss_finalize<<<1, 64, 0, stream>>>(out, ws);
}